// SelfAttention_12017318494501
// MI455X (gfx1250) — compile-verified
//
#include <hip/hip_runtime.h>

typedef float v2f __attribute__((ext_vector_type(2)));
typedef float v4f __attribute__((ext_vector_type(4)));
typedef float v8f __attribute__((ext_vector_type(8)));
typedef int   v4i __attribute__((ext_vector_type(4)));

#define SEQ    128
#define ISZ    512
#define STILE  16
#define NTILES (SEQ / STILE)
#define ROWPAD 516              // 516 % 64 == 4 -> conflict-free A-frag gathers
#define WAVES  4

#if __has_builtin(__builtin_amdgcn_wmma_f32_16x16x4_f32)
#define HAS_WMMA4 1
#endif

__device__ __forceinline__ void ld_async_b128(const float* g, float* l) {
#if __has_builtin(__builtin_amdgcn_global_load_async_to_lds_b128)
  typedef __attribute__((address_space(1))) v4i* gp_t;
  typedef __attribute__((address_space(3))) v4i* lp_t;
  __builtin_amdgcn_global_load_async_to_lds_b128(
      (gp_t)(unsigned long long)g,
      (lp_t)(unsigned)(unsigned long long)l, 0, 0);
#else
  unsigned lofs = (unsigned)(unsigned long long)l;  // generic LDS ptr low 32 bits = LDS offset
  asm volatile("global_load_async_to_lds_b128 %0, %1, off" :: "v"(lofs), "v"(g) : "memory");
#endif
}

__device__ __forceinline__ void wait_async0() {
#if __has_builtin(__builtin_amdgcn_s_wait_asynccnt)
  __builtin_amdgcn_s_wait_asynccnt(0);
#else
  asm volatile("s_wait_asynccnt 0" ::: "memory");
#endif
}

__device__ __forceinline__ void wait_ds0() {
#if __has_builtin(__builtin_amdgcn_s_wait_dscnt)
  __builtin_amdgcn_s_wait_dscnt(0);
#else
  asm volatile("s_wait_dscnt 0" ::: "memory");
#endif
}

__device__ __forceinline__ float fast_tanh(float v) {
#if __has_builtin(__builtin_amdgcn_tanhf)
  return __builtin_amdgcn_tanhf(v);      // V_TANH_F32 (CDNA5 trans op)
#else
  return tanhf(v);
#endif
}

__device__ __forceinline__ float lane_bcast(float v, int srcLane) {
  return __builtin_bit_cast(float,
      __builtin_amdgcn_readlane(__builtin_bit_cast(int, v), srcLane));
}

__global__ __launch_bounds__(WAVES * 32)
void attn_tanh_pool_kernel(const float* __restrict__ x,
                           const float* __restrict__ w,
                           const float* __restrict__ bias,
                           float* __restrict__ out,
                           int batches) {
  __shared__ float smem[WAVES][STILE][ROWPAD];   // 132096 B, per-wave private buffers
  const int lane = threadIdx.x & 31;
  const int wave = threadIdx.x >> 5;
  const int b    = blockIdx.x * WAVES + wave;
  if (b >= batches) return;                      // wave-uniform (batches % WAVES == 0)

  float* buf = &smem[wave][0][0];
  const float* xb = x + (size_t)b * (SEQ * ISZ);

  const int mm  = lane & 15;                     // WMMA M (=s within tile) / N index
  const int khi = (lane >> 4) * 2;               // K sub-offset per half-wave

  // out element ownership: i = 4*lane + 128*q + c  (full-bandwidth b128 LDS reads)
  v4f acc0 = {0.f,0.f,0.f,0.f}, acc1 = acc0, acc2 = acc0, acc3 = acc0;

  for (int t = 0; t < NTILES; ++t) {
    const int s0 = t * STILE;

    // ---- stream x[b, s0:s0+16, :] into this wave's LDS buffer (async, 64x b128) ----
    wait_ds0();                                  // prior tile's ds_loads retired before overwrite
    {
      const float* gbase = xb + (size_t)s0 * ISZ;
      #pragma unroll
      for (int r = 0; r < STILE; ++r) {
        #pragma unroll
        for (int cc = 0; cc < 4; ++cc) {
          const int col = (cc * 32 + lane) * 4;
          ld_async_b128(gbase + r * ISZ + col, buf + r * ROWPAD + col);
        }
      }
    }
    wait_async0();

    // ---- phase 1: 16 scores via V_WMMA_F32_16X16X4_F32 diagonal ----
    float as[STILE];
    {
#ifdef HAS_WMMA4
      v8f d0 = {0.f,0.f,0.f,0.f,0.f,0.f,0.f,0.f};
      v8f d1 = d0;
      const float* arow = buf + mm * ROWPAD + khi;                 // LDS (A-frag row)
      const float* wrow = w + (size_t)(s0 + mm) * ISZ + khi;       // global (B-frag row, L2-hot)
      #pragma unroll 8
      for (int kk = 0; kk < ISZ; kk += 8) {
        v2f a0 = *(const v2f*)(arow + kk);
        v2f b0 = *(const v2f*)(wrow + kk);
        v2f a1 = *(const v2f*)(arow + kk + 4);
        v2f b1 = *(const v2f*)(wrow + kk + 4);
        d0 = __builtin_amdgcn_wmma_f32_16x16x4_f32(false, a0, false, b0,
                                                   (short)0, d0, false, false);
        d1 = __builtin_amdgcn_wmma_f32_16x16x4_f32(false, a1, false, b1,
                                                   (short)0, d1, false, false);
      }
      // diag(D): s=r at (vgpr r, lane r); s=8+r at (vgpr r, lane 24+r)
      #pragma unroll
      for (int r = 0; r < 8; ++r) {
        float v  = d0[r] + d1[r];
        float lo = lane_bcast(v, r);
        float hi = lane_bcast(v, 24 + r);
        as[r]     = fast_tanh(lo + bias[s0 + r]);
        as[r + 8] = fast_tanh(hi + bias[s0 + r + 8]);
      }
#else
      // fallback: per-s shuffle reduction
      #pragma unroll
      for (int s = 0; s < STILE; ++s) {
        const float* row = buf + s * ROWPAD + 4 * lane;
        const float* wr  = w + (size_t)(s0 + s) * ISZ + 4 * lane;
        float p = 0.f;
        #pragma unroll
        for (int q = 0; q < 4; ++q) {
          v4f xv = *(const v4f*)(row + 128 * q);
          v4f wv = *(const v4f*)(wr + 128 * q);
          p += xv.x * wv.x + xv.y * wv.y + xv.z * wv.z + xv.w * wv.w;
        }
        #pragma unroll
        for (int off = 16; off > 0; off >>= 1) p += __shfl_xor(p, off, 32);
        as[s] = fast_tanh(p + bias[s0 + s]);
      }
#endif
    }

    // ---- phase 2: acc += a[s] * x[b, s, :] from the LDS-resident tile ----
    #pragma unroll
    for (int s = 0; s < STILE; ++s) {
      const float* row = buf + s * ROWPAD + 4 * lane;
      v4f x0 = *(const v4f*)(row);
      v4f x1 = *(const v4f*)(row + 128);
      v4f x2 = *(const v4f*)(row + 256);
      v4f x3 = *(const v4f*)(row + 384);
      const float a = as[s];
      acc0 += x0 * a; acc1 += x1 * a; acc2 += x2 * a; acc3 += x3 * a;
    }
  }

  float* ob = out + (size_t)b * ISZ + 4 * lane;
  *(v4f*)(ob)       = acc0;
  *(v4f*)(ob + 128) = acc1;
  *(v4f*)(ob + 256) = acc2;
  *(v4f*)(ob + 384) = acc3;
}

extern "C" void kernel_launch(void* const* d_in, const int* in_sizes, int n_in,
                              void* d_out, int out_size, void* d_ws, size_t ws_size,
                              hipStream_t stream) {
  const float* x    = (const float*)d_in[0];   // [B, 128, 512] f32
  const float* w    = (const float*)d_in[1];   // [128, 512, 1] f32
  const float* bias = (const float*)d_in[2];   // [128, 1, 1]  f32
  float* out        = (float*)d_out;           // [B, 512] f32

  const int batches = in_sizes[0] / (SEQ * ISZ);
  const int blocks  = (batches + WAVES - 1) / WAVES;
  attn_tanh_pool_kernel<<<blocks, WAVES * 32, 0, stream>>>(x, w, bias, out, batches);
}